// NonstationaryGaussianKernel_19224273617254
// MI455X (gfx1250) — compile-verified
//
#include <hip/hip_runtime.h>

#define NPTS 8192
#define LAT  64

typedef __attribute__((ext_vector_type(2))) float v2f;
typedef __attribute__((ext_vector_type(8))) float v8f;

__device__ __forceinline__ float fast_exp(float z) {
    return __builtin_amdgcn_exp2f(z * 1.44269504088896340736f);
}
__device__ __forceinline__ float fast_ln(float z) {
    return __builtin_amdgcn_logf(z) * 0.69314718055994530942f;
}

// Stage 1: per-point lengthscale MLP (softplus(W2 @ selu(W1 p + b1) + b2))
// plus squared norm.  t < NPTS -> x points, else y points.
__global__ __launch_bounds__(256) void nsgk_scales(
    const float* __restrict__ x, const float* __restrict__ y,
    const float* __restrict__ W1, const float* __restrict__ b1,
    const float* __restrict__ W2, const float* __restrict__ b2,
    float* __restrict__ sx, float* __restrict__ sy,
    float* __restrict__ xx, float* __restrict__ yy)
{
    int t = blockIdx.x * blockDim.x + threadIdx.x;
    if (t >= 2 * NPTS) return;
    const float* p = (t < NPTS) ? (x + 3 * t) : (y + 3 * (t - NPTS));
    float p0 = p[0], p1 = p[1], p2 = p[2];

    const float kScale = 1.0507009873554804934f;   // jax.nn.selu scale
    const float kAlpha = 1.6732632423543772848f;   // jax.nn.selu alpha

    float acc = b2[0];
#pragma unroll 8
    for (int j = 0; j < LAT; ++j) {
        float z = W1[3*j+0]*p0 + W1[3*j+1]*p1 + W1[3*j+2]*p2 + b1[j];
        float h = (z > 0.0f) ? (kScale * z)
                             : (kScale * kAlpha * (fast_exp(z) - 1.0f));
        acc += h * W2[j];
    }
    // softplus, numerically safe for large positive input
    float s = (acc > 15.0f) ? acc : fast_ln(1.0f + fast_exp(acc));
    float nrm2 = p0*p0 + p1*p1 + p2*p2;

    if (t < NPTS) { sx[t] = s;        xx[t] = nrm2; }
    else          { sy[t-NPTS] = s;   yy[t-NPTS] = nrm2; }
}

// Stage 2: pairwise kernel matrix.  One wave = one 16-row tile, sweeps 8
// column tiles.  x.y^T (K=3 zero-padded to 4) via V_WMMA_F32_16X16X4_F32.
//
// A (16x4 f32, 2 VGPRs): lanes 0-15 hold {K=0,K=1}, lanes 16-31 {K=2,K=3}.
// B (4x16 f32) mirrors that striping with N = lane%16.
// D (16x16 f32, 8 VGPRs): element (M = r + 8*(lane/16), N = lane%16) in reg r.
__global__ __launch_bounds__(256) void nsgk_pairwise(
    const float* __restrict__ x,  const float* __restrict__ y,
    const float* __restrict__ sx, const float* __restrict__ sy,
    const float* __restrict__ xx, const float* __restrict__ yy,
    float* __restrict__ out)
{
    const int lane = threadIdx.x & 31;
    const int wave = threadIdx.x >> 5;
    const int half = lane >> 4;        // 0: lanes 0-15, 1: lanes 16-31
    const int mi   = lane & 15;

    const int rt  = blockIdx.y * 8 + wave;   // row tile   [0,512)
    const int ct0 = blockIdx.x * 8;          // first col tile of this block

    // A operand: row (rt*16 + mi), K-slice selected by half; K=3 padded to 0.
    const int rowA = rt * 16 + mi;
    v2f a;
    a.x = half ? x[3*rowA + 2] : x[3*rowA + 0];
    a.y = half ? 0.0f          : x[3*rowA + 1];

    // Row-side epilogue values for the 8 D registers this lane owns.
    float sxv[8], xxv[8];
#pragma unroll
    for (int r = 0; r < 8; ++r) {
        int m = rt * 16 + 8 * half + r;
        sxv[r] = sx[m];
        xxv[r] = xx[m];
    }

    for (int c8 = 0; c8 < 8; ++c8) {
        const int ct   = ct0 + c8;
        const int colB = ct * 16 + mi;       // both B-load and epilogue column

        v2f b;
        b.x = half ? y[3*colB + 2] : y[3*colB + 0];
        b.y = half ? 0.0f          : y[3*colB + 1];

        const float syv = sy[colB];
        const float yyv = yy[colB];
        const float sy2 = syv * syv;

        v8f acc = {};
        acc = __builtin_amdgcn_wmma_f32_16x16x4_f32(
            /*neg_a=*/false, a, /*neg_b=*/false, b,
            /*c_mod=*/(short)0, acc, /*reuse_a=*/false, /*reuse_b=*/false);

#pragma unroll
        for (int r = 0; r < 8; ++r) {
            float dot   = acc[r];                        // x_m . y_n
            float d2    = xxv[r] + yyv - 2.0f * dot;     // |x-y|^2
            float denom = sxv[r] * sxv[r] + sy2;
            float rden  = __builtin_amdgcn_rcpf(denom);
            float amp   = __builtin_amdgcn_sqrtf(2.0f * sxv[r] * syv * rden);
            float e     = __builtin_amdgcn_exp2f(-d2 * rden * 1.44269504088896340736f);
            int   row   = rt * 16 + 8 * half + r;
            out[(size_t)row * NPTS + colB] = amp * e;
        }
    }
}

extern "C" void kernel_launch(void* const* d_in, const int* in_sizes, int n_in,
                              void* d_out, int out_size, void* d_ws, size_t ws_size,
                              hipStream_t stream)
{
    const float* x  = (const float*)d_in[0];
    const float* y  = (const float*)d_in[1];
    const float* W1 = (const float*)d_in[2];
    const float* b1 = (const float*)d_in[3];
    const float* W2 = (const float*)d_in[4];
    const float* b2 = (const float*)d_in[5];
    float* out = (float*)d_out;

    float* ws = (float*)d_ws;          // 4 * 8192 floats = 128 KB scratch
    float* sx = ws;
    float* sy = ws + NPTS;
    float* xx = ws + 2 * NPTS;
    float* yy = ws + 3 * NPTS;

    nsgk_scales<<<(2 * NPTS + 255) / 256, 256, 0, stream>>>(
        x, y, W1, b1, W2, b2, sx, sy, xx, yy);

    dim3 grid(64, 64);                 // 64*8=512 col tiles, 64*8=512 row tiles
    nsgk_pairwise<<<grid, 256, 0, stream>>>(x, y, sx, sy, xx, yy, out);
}